// _BatchOnlyModel_60919816127158
// MI455X (gfx1250) — compile-verified
//
#include <hip/hip_runtime.h>
#include <hip/hip_bf16.h>

// Segment-mean of x[:,0] over sorted int32 segment ids b into n_graphs outputs.
// Strategy: per-block LDS aggregation (keys are sorted -> tiny per-chunk key
// range), NT loads for the strided column, gfx1250 global_prefetch for latency
// hiding, then a handful of global float atomics per block.

#define THREADS 256
#define NPT 8
#define CHUNK (THREADS * NPT)   // 2048 nodes per block
#define NFEAT 64
#define MAX_G 4096              // n_graphs upper bound (reference: 4096)

__global__ void seg_zero_kernel(float* __restrict__ ws, int count) {
    int i = blockIdx.x * blockDim.x + threadIdx.x;
    if (i < count) ws[i] = 0.0f;
}

__global__ void seg_accum_kernel(const float* __restrict__ x,
                                 const int*   __restrict__ b,
                                 float* __restrict__ sums,
                                 float* __restrict__ cnts,
                                 int n) {
    __shared__ float lsum[MAX_G];
    __shared__ float lcnt[MAX_G];

    const int base = blockIdx.x * CHUNK;
    if (base >= n) return;
    const int end = (base + CHUNK < n) ? (base + CHUNK) : n;

    // b is sorted: the chunk's keys lie in [first, last]. Broadcast loads.
    const int first = b[base];
    const int last  = b[end - 1];
    const int range = last - first + 1;   // typically ~5, worst case MAX_G

    for (int i = threadIdx.x; i < range; i += THREADS) {
        lsum[i] = 0.0f;
        lcnt[i] = 0.0f;
    }
    __syncthreads();

    // Strided column gather + LDS atomic aggregation.
    #pragma unroll
    for (int k = 0; k < NPT; ++k) {
        const int i = base + k * THREADS + threadIdx.x;
        if (i < end) {
            // Prefetch next iteration's cacheline (speculative NT prefetch;
            // OOB translations are silently dropped per ISA).
            __builtin_prefetch(&x[(size_t)(i + THREADS) * NFEAT], 0, 0);
            // Non-temporal: 256 MB of dead-row traffic must not evict the
            // accumulator working set from L2.
            const float v = __builtin_nontemporal_load(&x[(size_t)i * NFEAT]);
            const int key = b[i] - first;
            atomicAdd(&lsum[key], v);      // ds_add_f32
            atomicAdd(&lcnt[key], 1.0f);   // ds_add_f32
        }
    }
    __syncthreads();

    // Flush touched segments with global float atomics (few per block).
    for (int i = threadIdx.x; i < range; i += THREADS) {
        const float c = lcnt[i];
        if (c != 0.0f) {
            atomicAdd(&sums[first + i], lsum[i]);   // global_atomic_add_f32
            atomicAdd(&cnts[first + i], c);
        }
    }
}

__global__ void seg_div_kernel(const float* __restrict__ sums,
                               const float* __restrict__ cnts,
                               float* __restrict__ out,
                               int g) {
    int i = blockIdx.x * blockDim.x + threadIdx.x;
    if (i < g) out[i] = sums[i] / cnts[i];   // 0/0 -> NaN, matching reference
}

extern "C" void kernel_launch(void* const* d_in, const int* in_sizes, int n_in,
                              void* d_out, int out_size, void* d_ws, size_t ws_size,
                              hipStream_t stream) {
    const float* x = (const float*)d_in[0];
    const int*   b = (const int*)d_in[1];
    float* out = (float*)d_out;

    const int n = in_sizes[1];     // number of nodes (2,000,000)
    const int g = out_size;        // n_graphs (4096), output is (g,1)

    float* sums = (float*)d_ws;
    float* cnts = sums + g;

    // 1) zero accumulators
    {
        int count = 2 * g;
        int blocks = (count + THREADS - 1) / THREADS;
        seg_zero_kernel<<<blocks, THREADS, 0, stream>>>(sums, count);
    }
    // 2) block-chunked LDS aggregation + global atomic flush
    {
        int blocks = (n + CHUNK - 1) / CHUNK;
        seg_accum_kernel<<<blocks, THREADS, 0, stream>>>(x, b, sums, cnts, n);
    }
    // 3) divide
    {
        int blocks = (g + THREADS - 1) / THREADS;
        seg_div_kernel<<<blocks, THREADS, 0, stream>>>(sums, cnts, out, g);
    }
}